// CustomContrastiveLoss_16149077033499
// MI455X (gfx1250) — compile-verified
//
#include <hip/hip_runtime.h>
#include <hip/hip_bf16.h>

// ---------------------------------------------------------------------------
// SupCon-style contrastive loss, fully fused for MI455X (gfx1250, wave32).
//   prep_kernel     : gather + JAX-threefry dropout + row L2-normalize
//   gram_kernel     : f32 WMMA Gram matrix fused with exp/mask row reduction
//   finalize_kernel : scalar loss
// ---------------------------------------------------------------------------

#define DFEAT      1024
#define KC         128
#define LDS_STRIDE 132          // 128 + 4 pad -> 4-bank rotation per row, float4-aligned
#define INV_TEMP   (1.0f / 0.07f)
#define EPS_NRM    1e-8f

typedef __attribute__((ext_vector_type(2))) float v2f;
typedef __attribute__((ext_vector_type(8))) float v8f;

#if defined(__HIP_DEVICE_COMPILE__)
#if !__has_builtin(__builtin_amdgcn_wmma_f32_16x16x4_f32)
#error "gfx1250: __builtin_amdgcn_wmma_f32_16x16x4_f32 not available on this toolchain"
#endif
#endif

// ------------------------- JAX threefry-2x32 -------------------------------
__device__ __forceinline__ unsigned rotl32(unsigned x, int r) {
    return (x << r) | (x >> (32 - r));
}

// Reproduces jax.random.bernoulli(jax.random.key(42), ...) bit stream:
// counters = iota(n); pairs (j, j+n/2); 20-round threefry with key (0, 42).
__device__ unsigned threefry_bits(unsigned j, unsigned n) {
    const unsigned h = n >> 1;
    unsigned x0, x1;
    int sel;
    if (j < h) { x0 = j;     x1 = j + h; sel = 0; }
    else       { x0 = j - h; x1 = j;     sel = 1; }
    const unsigned k0 = 0u, k1 = 42u, k2 = 0u ^ 42u ^ 0x1BD11BDAu;
    x0 += k0; x1 += k1;
#define TF_RND(r) { x0 += x1; x1 = rotl32(x1, r); x1 ^= x0; }
    TF_RND(13) TF_RND(15) TF_RND(26) TF_RND(6)
    x0 += k1; x1 += k2 + 1u;
    TF_RND(17) TF_RND(29) TF_RND(16) TF_RND(24)
    x0 += k2; x1 += k0 + 2u;
    TF_RND(13) TF_RND(15) TF_RND(26) TF_RND(6)
    x0 += k0; x1 += k1 + 3u;
    TF_RND(17) TF_RND(29) TF_RND(16) TF_RND(24)
    x0 += k1; x1 += k2 + 4u;
    TF_RND(13) TF_RND(15) TF_RND(26) TF_RND(6)
    x0 += k2; x1 += k0 + 5u;
#undef TF_RND
    return sel ? x1 : x0;
}

// ------------------------------ prep ---------------------------------------
// One 256-thread block per (padded) output row.
__global__ __launch_bounds__(256) void prep_kernel(
    const float* __restrict__ features, const int* __restrict__ labels,
    const int* __restrict__ aug_indices, int N, int n_aug, int N2,
    float* __restrict__ fn, int* __restrict__ labels2,
    float* __restrict__ diag, float* __restrict__ denom)
{
    const int i   = blockIdx.x;
    const int tid = threadIdx.x;
    __shared__ float red[256];
    __shared__ float s_inv;

    float x[4];
    int lab = -1;
    if (i < N) {
        const float* src = features + (size_t)i * DFEAT;
#pragma unroll
        for (int k = 0; k < 4; ++k) x[k] = src[tid + k * 256];
        lab = labels[i];
    } else if (i < N2) {
        const int a  = i - N;
        const int gi = aug_indices[a];
        const float* src = features + (size_t)gi * DFEAT;
        const unsigned n = (unsigned)n_aug * (unsigned)DFEAT;
#pragma unroll
        for (int k = 0; k < 4; ++k) {
            const int d = tid + k * 256;
            const float v = src[d];
            const unsigned bits = threefry_bits((unsigned)a * (unsigned)DFEAT + (unsigned)d, n);
            const float u = __uint_as_float(0x3f800000u | (bits >> 9)) - 1.0f;
            x[k] = (u < 0.5f) ? v * 2.0f : 0.0f;   // dropout p=0.5, scale 1/(1-p)=2
        }
        lab = labels[gi];
    } else {                                        // pad rows (to multiple of 128)
        x[0] = x[1] = x[2] = x[3] = 0.0f;
        lab = -1;
    }

    float ss = x[0]*x[0] + x[1]*x[1] + x[2]*x[2] + x[3]*x[3];
    red[tid] = ss;
    __syncthreads();
    for (int s = 128; s > 0; s >>= 1) {
        if (tid < s) red[tid] += red[tid + s];
        __syncthreads();
    }
    if (tid == 0) {
        const float sumsq = red[0];
        const float inv   = 1.0f / fmaxf(sqrtf(sumsq), EPS_NRM);
        s_inv      = inv;
        diag[i]    = sumsq * inv * inv * INV_TEMP;  // sim_ii
        denom[i]   = 0.0f;                          // re-zeroed every launch
        labels2[i] = lab;
    }
    __syncthreads();
    const float inv = s_inv;
    float* dst = fn + (size_t)i * DFEAT;
#pragma unroll
    for (int k = 0; k < 4; ++k) dst[tid + k * 256] = x[k] * inv;
}

// ------------------------------ gram ---------------------------------------
// Workgroup = 64-row block, 8 waves. Each pass covers 128 columns (16/wave).
// A K-chunk staged in LDS (shared by all waves), B streamed from L2.
// Per k-step: 4x v_wmma_f32_16x16x4_f32 (exact f32, matches reference dtype).
__global__ __launch_bounds__(256) void gram_kernel(
    const float* __restrict__ fn, const int* __restrict__ labels2,
    float* __restrict__ denom, int N2pad)
{
    __shared__ float ldsA[64 * LDS_STRIDE];
    __shared__ int   s_lrow[64];

    const int tid     = threadIdx.x;
    const int lane    = tid & 31;
    const int wv      = tid >> 5;
    const int lane15  = lane & 15;
    const int koff    = (lane >> 4) * 2;   // A/B fragment K-offset per lane half
    const int halfOff = (lane >> 4) * 8;   // C/D row offset per lane half
    const int RB      = blockIdx.x * 64;

    if (tid < 64) s_lrow[tid] = labels2[RB + tid];

    float dacc[4][8];
#pragma unroll
    for (int t = 0; t < 4; ++t)
#pragma unroll
        for (int r = 0; r < 8; ++r) dacc[t][r] = 0.0f;

    const int nGroups = N2pad >> 7;
    for (int g = 0; g < nGroups; ++g) {
        const int J = (g << 7) + wv * 16;
        v8f c[4];
#pragma unroll
        for (int t = 0; t < 4; ++t) c[t] = (v8f)0.0f;

        const float* bptr = fn + (size_t)(J + lane15) * DFEAT + koff;

        for (int kc = 0; kc < DFEAT; kc += KC) {
            __syncthreads();
            // cooperative stage of A[64][KC] into LDS (float4, conflict-free pad)
#pragma unroll
            for (int it = 0; it < 8; ++it) {
                const int e   = it * 256 + tid;
                const int row = e >> 5;
                const int c4  = (e & 31) << 2;
                const float4 v = *reinterpret_cast<const float4*>(
                    fn + (size_t)(RB + row) * DFEAT + kc + c4);
                *reinterpret_cast<float4*>(&ldsA[row * LDS_STRIDE + c4]) = v;
            }
            __syncthreads();

#pragma unroll 4
            for (int kk = 0; kk < KC; kk += 4) {
                const v2f b = *reinterpret_cast<const v2f*>(bptr + kc + kk);
#pragma unroll
                for (int t = 0; t < 4; ++t) {
                    const v2f a = *reinterpret_cast<const v2f*>(
                        &ldsA[(t * 16 + lane15) * LDS_STRIDE + kk + koff]);
                    c[t] = __builtin_amdgcn_wmma_f32_16x16x4_f32(
                        false, a, false, b, (short)0, c[t], false, false);
                }
            }
        }

        // fused epilogue: sim = dot/T; accumulate exp(sim)*[labels match]
        const int labc = labels2[J + lane15];
#pragma unroll
        for (int t = 0; t < 4; ++t)
#pragma unroll
            for (int r = 0; r < 8; ++r) {
                const int labr = s_lrow[t * 16 + r + halfOff];
                const float s  = c[t][r] * INV_TEMP;
                dacc[t][r] += (labr == labc) ? __expf(s) : 0.0f;
            }
    }

    // reduce 16 columns per lane-half, then one atomic per (row) per wave
#pragma unroll
    for (int t = 0; t < 4; ++t)
#pragma unroll
        for (int r = 0; r < 8; ++r) {
            float v = dacc[t][r];
            v += __shfl_xor(v, 1, 32);
            v += __shfl_xor(v, 2, 32);
            v += __shfl_xor(v, 4, 32);
            v += __shfl_xor(v, 8, 32);
            if (lane15 == 0)
                atomicAdd(&denom[RB + t * 16 + r + halfOff], v);
        }
}

// ---------------------------- finalize -------------------------------------
__global__ __launch_bounds__(256) void finalize_kernel(
    const float* __restrict__ diag, const float* __restrict__ denom,
    int N2, float* __restrict__ out)
{
    __shared__ float red[256];
    const int tid = threadIdx.x;
    float acc = 0.0f;
    for (int i = tid; i < N2; i += 256) acc += diag[i] - logf(denom[i]);
    red[tid] = acc;
    __syncthreads();
    for (int s = 128; s > 0; s >>= 1) {
        if (tid < s) red[tid] += red[tid + s];
        __syncthreads();
    }
    if (tid == 0) out[0] = -red[0] / (float)N2;
}

// ---------------------------------------------------------------------------
extern "C" void kernel_launch(void* const* d_in, const int* in_sizes, int n_in,
                              void* d_out, int out_size, void* d_ws, size_t ws_size,
                              hipStream_t stream) {
    const float* features    = (const float*)d_in[0];
    const int*   labels      = (const int*)d_in[1];
    const int*   aug_indices = (const int*)d_in[2];

    const int N     = in_sizes[1];            // 6144
    const int n_aug = in_sizes[2];
    const int N2    = N + n_aug;
    const int N2pad = (N2 + 127) & ~127;      // multiple of 128 (col group & row block)

    char* w = (char*)d_ws;
    float* fn      = (float*)w; w += (size_t)N2pad * DFEAT * sizeof(float);
    int*   labels2 = (int*)w;   w += (size_t)N2pad * sizeof(int);
    float* diag    = (float*)w; w += (size_t)N2pad * sizeof(float);
    float* denom   = (float*)w; w += (size_t)N2pad * sizeof(float);

    prep_kernel<<<N2pad, 256, 0, stream>>>(features, labels, aug_indices,
                                           N, n_aug, N2, fn, labels2, diag, denom);
    gram_kernel<<<N2pad / 64, 256, 0, stream>>>(fn, labels2, denom, N2pad);
    finalize_kernel<<<1, 256, 0, stream>>>(diag, denom, N2, (float*)d_out);
}